// HashGridEncoder_13451837571304
// MI455X (gfx1250) — compile-verified
//
#include <hip/hip_runtime.h>
#include <math.h>
#include <stdint.h>

#define NLEVELS 16
#define STAGED  3          // levels whose tables are staged in LDS (4096+8000+15625 entries = 216.6 KB < 320 KB/WG)
#define BLOCK   1024       // 32 wave32 waves per workgroup

typedef __attribute__((ext_vector_type(4))) float v4f;
typedef __attribute__((ext_vector_type(2))) int   v2i;   // pointee type the async-LDS builtin expects (b64)

struct Params {
  const float2* tbl[NLEVELS];
  float    resf[NLEVELS];
  uint32_t res[NLEVELS];
  uint32_t size[NLEVELS];
  uint32_t magic[NLEVELS];     // floor(2^32 / size) for Barrett mod
  uint32_t ldsBase[STAGED];    // in float2 units
  uint32_t stagedEntries[STAGED];
};

// gfx1250 async global->LDS copy path (ASYNCcnt), guarded so compile can't break.
#if defined(__HIP_DEVICE_COMPILE__) && __has_builtin(__builtin_amdgcn_global_load_async_to_lds_b64)
#define HAS_ASYNC_LDS 1
#else
#define HAS_ASYNC_LDS 0
#endif

#if HAS_ASYNC_LDS
typedef __attribute__((address_space(1))) v2i* gv2i_p;   // global (device) address space
typedef __attribute__((address_space(3))) v2i* lv2i_p;   // LDS address space
#endif

// Exact h % d via Barrett: M = floor(2^32/d). q = umulhi(h,M) is in {q_true-1, q_true},
// so one conditional correction suffices. For power-of-two d (2^19 levels) it is exact directly.
__device__ __forceinline__ uint32_t fastmod(uint32_t h, uint32_t d, uint32_t M) {
  uint32_t q = __umulhi(h, M);
  uint32_t r = h - q * d;
  return (r >= d) ? (r - d) : r;
}

__global__ __launch_bounds__(BLOCK) void hashgrid_encode(
    const float* __restrict__ x, float* __restrict__ out, int n, Params p) {
  extern __shared__ float2 sh[];

  // ---- Stage small low-level tables into LDS (streaming copy replaces random L2 gathers) ----
  for (int l = 0; l < STAGED; ++l) {
    const float2* t = p.tbl[l];
    const uint32_t base = p.ldsBase[l];
    const uint32_t sz   = p.stagedEntries[l];
    for (uint32_t i = threadIdx.x; i < sz; i += BLOCK) {
#if HAS_ASYNC_LDS
      __builtin_amdgcn_global_load_async_to_lds_b64(
          (gv2i_p)(t + i),
          (lv2i_p)(sh + base + i),
          0, 0);
#else
      sh[base + i] = t[i];
#endif
    }
  }
#if HAS_ASYNC_LDS
#if __has_builtin(__builtin_amdgcn_s_wait_asynccnt)
  __builtin_amdgcn_s_wait_asynccnt(0);
#else
  asm volatile("s_wait_asynccnt 0" ::: "memory");
#endif
#endif
  __syncthreads();

  const int pt = blockIdx.x * BLOCK + threadIdx.x;
  if (pt >= n) return;

  // x is read-once: non-temporal so it never competes with the L2-resident tables.
  const float* xp = x + (size_t)pt * 3u;
  const float xv = __builtin_nontemporal_load(xp + 0);
  const float yv = __builtin_nontemporal_load(xp + 1);
  const float zv = __builtin_nontemporal_load(xp + 2);

  float acc[2 * NLEVELS];

#pragma unroll
  for (int l = 0; l < NLEVELS; ++l) {
    const float    rf  = p.resf[l];
    const uint32_t res = p.res[l];
    const uint32_t d   = p.size[l];
    const uint32_t M   = p.magic[l];

    const float px = xv * rf, py = yv * rf, pz = zv * rf;
    const float fxf = floorf(px), fyf = floorf(py), fzf = floorf(pz);
    const float wx = px - fxf, wy = py - fyf, wz = pz - fzf;

    const uint32_t ix0 = (uint32_t)(int)fxf;
    const uint32_t iy0 = (uint32_t)(int)fyf;
    const uint32_t iz0 = (uint32_t)(int)fzf;
    uint32_t ix1 = ix0 + 1u; ix1 = (ix1 == res) ? 0u : ix1;   // (p0+1) mod res
    uint32_t iy1 = iy0 + 1u; iy1 = (iy1 == res) ? 0u : iy1;
    uint32_t iz1 = iz0 + 1u; iz1 = (iz1 == res) ? 0u : iz1;

    // primes: dim0 * 1, dim1 * 2654435761, dim2 * 805459861  (uint32 wrap == reference)
    const uint32_t a0 = ix0,                b0 = iy0 * 2654435761u, c0 = iz0 * 805459861u;
    const uint32_t a1 = ix1,                b1 = iy1 * 2654435761u, c1 = iz1 * 805459861u;

    const float u0 = 1.0f - wx, u1 = wx;
    const float v0 = 1.0f - wy, v1 = wy;
    const float s0 = 1.0f - wz, s1 = wz;

    float f0 = 0.0f, f1 = 0.0f;
    const float2* t = p.tbl[l];

#pragma unroll
    for (int c = 0; c < 8; ++c) {   // bit2->dim0, bit1->dim1, bit0->dim2 (matches CORNER_BITS)
      const uint32_t h = fastmod(((c & 4) ? a1 : a0) +
                                 ((c & 2) ? b1 : b0) +
                                 ((c & 1) ? c1 : c0), d, M);
      const float2 f = (l < STAGED) ? sh[p.ldsBase[l] + h] : t[h];
      const float  w = ((c & 4) ? u1 : u0) * ((c & 2) ? v1 : v0) * ((c & 1) ? s1 : s0);
      f0 = fmaf(f.x, w, f0);
      f1 = fmaf(f.y, w, f1);
    }
    acc[2 * l]     = f0;
    acc[2 * l + 1] = f1;
  }

  // 256 MB streaming output: non-temporal b128 stores so it doesn't evict tables from L2.
  float* ob = out + (size_t)pt * 32u;
#pragma unroll
  for (int k = 0; k < 8; ++k) {
    v4f v = { acc[4 * k + 0], acc[4 * k + 1], acc[4 * k + 2], acc[4 * k + 3] };
    __builtin_nontemporal_store(v, (v4f*)(ob + 4 * k));
  }
}

extern "C" void kernel_launch(void* const* d_in, const int* in_sizes, int n_in,
                              void* d_out, int out_size, void* d_ws, size_t ws_size,
                              hipStream_t stream) {
  (void)n_in; (void)out_size; (void)d_ws; (void)ws_size;
  const float* x = (const float*)d_in[0];
  float* out = (float*)d_out;
  const int N = in_sizes[0] / 3;

  Params p;
  const double b = exp((log(512.0) - log(16.0)) / 15.0);
  for (int i = 0; i < NLEVELS; ++i) {
    long r = (long)(16.0 * pow(b, (double)i));             // same formula as reference
    const uint32_t size = (uint32_t)(in_sizes[1 + i] / 2); // exact table size from harness
    if (size < (1u << 19)) {                               // low levels: size == r^3, self-correct r
      while ((uint32_t)(r * r * r) < size) ++r;
      while ((uint32_t)(r * r * r) > size) --r;
    }
    p.tbl[i]   = (const float2*)d_in[1 + i];
    p.resf[i]  = (float)r;
    p.res[i]   = (uint32_t)r;
    p.size[i]  = size;
    p.magic[i] = (uint32_t)(0x100000000ULL / (uint64_t)size);
  }
  uint32_t total = 0;
  for (int i = 0; i < STAGED; ++i) {
    p.ldsBase[i] = total;
    p.stagedEntries[i] = p.size[i];
    total += p.size[i];
  }
  const size_t shbytes = (size_t)total * sizeof(float2);   // ~216.6 KB (<= 320 KB/WG on CDNA5)
  const int blocks = (N + BLOCK - 1) / BLOCK;

  hipLaunchKernelGGL(hashgrid_encode, dim3(blocks), dim3(BLOCK), shbytes, stream,
                     x, out, N, p);
}